// Color_Attention_86870008529139
// MI455X (gfx1250) — compile-verified
//
#include <hip/hip_runtime.h>
#include <hip/hip_bf16.h>
#include <math.h>

// ---------------------------------------------------------------------------
// Color (transposed/channel) attention, restructured for MI455X:
//   attn math via per-batch 192x192 covariances (q/k/v never materialized),
//   output = (Wo * blockdiag(softmax)) * Wv  folded into one 192x192 matrix,
//   then a single WMMA GEMM over the 16384 spatial positions.
// HBM traffic ~400 MB total @ 23.3 TB/s; compute ~39 GFLOP f32 WMMA.
// ---------------------------------------------------------------------------

#define DIM   192
#define HW    16384
#define NB    8
#define NHEAD 8
#define DH    24
#define EPS   1e-12f

typedef float v2f __attribute__((ext_vector_type(2)));
typedef float v8f __attribute__((ext_vector_type(8)));
typedef int   v4i __attribute__((ext_vector_type(4)));

// CDNA5 async global->LDS path (ASYNCcnt-tracked), if the toolchain exposes it
#if defined(__has_builtin)
#if __has_builtin(__builtin_amdgcn_global_load_async_to_lds_b128) && \
    __has_builtin(__builtin_amdgcn_s_wait_asynccnt)
#define USE_ASYNC_LDS 1
#endif
#endif

#ifdef USE_ASYNC_LDS
// per hipcc diagnostic: param0 is 'int __vector(4) __device__ *'  (AS1, non-const)
// param1 symmetric on the LDS side: 'int __vector(4) __shared__ *' (AS3)
typedef __attribute__((address_space(1))) v4i* gp4_t;
typedef __attribute__((address_space(3))) v4i* lp4_t;
#endif

// ============================================================
// Kernel 1: per-batch covariance  G = A * B^T  (192x192, K=16384)
// grid (18, 8), block 256 (8 waves). Each wave owns one 16x16 tile
// of a 32x64 block super-tile; K double-buffered through LDS in
// 64-wide chunks via async global->LDS copies (fallback: sync).
// ============================================================
__global__ __launch_bounds__(256)
void cov_kernel(const float* __restrict__ Abase,
                const float* __restrict__ Bbase,
                float* __restrict__ G)
{
    const int b      = blockIdx.y;
    const int m_base = (blockIdx.x % 6) * 32;   // 6 super-rows of 32
    const int n_base = (blockIdx.x / 6) * 64;   // 3 super-cols of 64

    const float* A  = Abase + (size_t)b * DIM * HW;
    const float* Bm = Bbase + (size_t)b * DIM * HW;

    const int tid  = threadIdx.x;
    const int lane = tid & 31;
    const int wave = tid >> 5;
    const int mt   = wave >> 2;      // 0..1
    const int nt   = wave & 3;       // 0..3

    const int ln = lane & 15;
    const int kh = lane >> 4;        // 0 or 1

    v8f acc = {};

#ifdef USE_ASYNC_LDS
    // ---------------- async double-buffered path ----------------
    // 68-float row pitch: row stride = 4 banks -> conflict-free lane access
    __shared__ float sA[2][32 * 68];
    __shared__ float sB[2][64 * 68];

    // per-thread 16B copy slots (A: 2 chunks, B: 4 chunks)
    const int rA0 = (tid      ) >> 4, cA0 = ((tid      ) & 15) << 2;
    const int rA1 = (tid + 256) >> 4, cA1 = ((tid + 256) & 15) << 2;

    auto issue = [&](int buf, int kc) {
        __builtin_amdgcn_global_load_async_to_lds_b128(
            (gp4_t)&A[(size_t)(m_base + rA0) * HW + kc + cA0],
            (lp4_t)&sA[buf][rA0 * 68 + cA0], 0, 0);
        __builtin_amdgcn_global_load_async_to_lds_b128(
            (gp4_t)&A[(size_t)(m_base + rA1) * HW + kc + cA1],
            (lp4_t)&sA[buf][rA1 * 68 + cA1], 0, 0);
        #pragma unroll
        for (int i = 0; i < 4; ++i) {
            int idx = tid + i * 256;              // 0..1023
            int row = idx >> 4;
            int c4  = (idx & 15) << 2;
            __builtin_amdgcn_global_load_async_to_lds_b128(
                (gp4_t)&Bm[(size_t)(n_base + row) * HW + kc + c4],
                (lp4_t)&sB[buf][row * 68 + c4], 0, 0);
        }
    };

    issue(0, 0);
    for (int kc = 0; kc < HW; kc += 64) {
        const int buf = (kc >> 6) & 1;
        __builtin_amdgcn_s_wait_asynccnt(0);  // this wave's copies landed
        __syncthreads();                      // all waves' copies landed;
                                              // prev-chunk compute finished
        if (kc + 64 < HW) issue(buf ^ 1, kc + 64);

        const float* aRow = &sA[buf][(mt * 16 + ln) * 68];
        const float* bRow = &sB[buf][(nt * 16 + ln) * 68];
        #pragma unroll
        for (int kk = 0; kk < 64; kk += 4) {
            // A frag 16x4: lane=m, K-half = lane>=16 -> 2 consecutive floats
            v2f a = *reinterpret_cast<const v2f*>(&aRow[kk + kh * 2]);
            // B frag 4x16: lane=n; vgpr0 holds K={0,1}, vgpr1 holds K={2,3}
            v2f bb;
            bb.x = bRow[kk + kh];
            bb.y = bRow[kk + kh + 2];
            acc = __builtin_amdgcn_wmma_f32_16x16x4_f32(
                false, a, false, bb, (short)0, acc, false, false);
        }
    }
#else
    // ---------------- synchronous fallback path ----------------
    __shared__ float sA[32 * 68];
    __shared__ float sB[64 * 68];

    for (int kc = 0; kc < HW; kc += 64) {
        __syncthreads();
        #pragma unroll
        for (int i = 0; i < 2; ++i) {
            int idx = tid + i * 256;              // 0..511
            int row = idx >> 4;
            int c4  = (idx & 15) << 2;
            float4 v = *reinterpret_cast<const float4*>(
                &A[(size_t)(m_base + row) * HW + kc + c4]);
            float* d = &sA[row * 68 + c4];
            d[0] = v.x; d[1] = v.y; d[2] = v.z; d[3] = v.w;
        }
        #pragma unroll
        for (int i = 0; i < 4; ++i) {
            int idx = tid + i * 256;              // 0..1023
            int row = idx >> 4;
            int c4  = (idx & 15) << 2;
            float4 v = *reinterpret_cast<const float4*>(
                &Bm[(size_t)(n_base + row) * HW + kc + c4]);
            float* d = &sB[row * 68 + c4];
            d[0] = v.x; d[1] = v.y; d[2] = v.z; d[3] = v.w;
        }
        __syncthreads();

        if (kc + 64 < HW) {
            __builtin_prefetch(&A[(size_t)(m_base + (tid >> 3)) * HW + kc + 64], 0, 0);
        }

        const float* aRow = &sA[(mt * 16 + ln) * 68];
        const float* bRow = &sB[(nt * 16 + ln) * 68];
        #pragma unroll
        for (int kk = 0; kk < 64; kk += 4) {
            v2f a = *reinterpret_cast<const v2f*>(&aRow[kk + kh * 2]);
            v2f bb;
            bb.x = bRow[kk + kh];
            bb.y = bRow[kk + kh + 2];
            acc = __builtin_amdgcn_wmma_f32_16x16x4_f32(
                false, a, false, bb, (short)0, acc, false, false);
        }
    }
#endif

    // ---- store 16x16 tile: D vgpr r -> row = r + 8*(lane>=16), col = lane&15
    float* Gb = G + (size_t)b * DIM * DIM;
    const int col   = n_base + nt * 16 + ln;
    const int rbase = m_base + mt * 16 + (kh << 3);
    #pragma unroll
    for (int r = 0; r < 8; ++r)
        Gb[(size_t)(rbase + r) * DIM + col] = acc[r];
}

// ============================================================
// Kernel 2: per (batch, head) attention math on covariances.
//   S = Wq_h * M * Wk_h^T ; norms from Cc/Ci ; softmax ;
//   Weff[:, h*24+d] = Wo[:, h*24+c] * P[c,d]
// grid (8 heads, 8 batches), block 256.
// ============================================================
__global__ __launch_bounds__(256)
void head_kernel(const float* __restrict__ M,
                 const float* __restrict__ Cc,
                 const float* __restrict__ Ci,
                 const float* __restrict__ Wq,
                 const float* __restrict__ Wk,
                 const float* __restrict__ Wo,
                 float* __restrict__ Weff)
{
    const int h   = blockIdx.x;
    const int b   = blockIdx.y;
    const int tid = threadIdx.x;
    const int h24 = h * DH;

    const float* Mb  = M  + (size_t)b * DIM * DIM;
    const float* Ccb = Cc + (size_t)b * DIM * DIM;
    const float* Cib = Ci + (size_t)b * DIM * DIM;

    __shared__ float sT[DH * DIM];      // 24x192 intermediate (reused)
    __shared__ float sS[DH * 25];       // raw scores
    __shared__ float sP[DH * 25];       // softmax probs
    __shared__ float sNq[DH], sNk[DH];  // inverse norms

    // T = Wq_h * M  (24x192)
    for (int idx = tid; idx < DH * DIM; idx += 256) {
        int c = idx / DIM, j = idx % DIM;
        const float* wr = &Wq[(h24 + c) * DIM];
        float acc = 0.f;
        for (int ci = 0; ci < DIM; ++ci) acc += wr[ci] * Mb[ci * DIM + j];
        sT[idx] = acc;
    }
    __syncthreads();
    // S = T * Wk_h^T  (24x24)
    for (int idx = tid; idx < DH * DH; idx += 256) {
        int c = idx / DH, d = idx % DH;
        const float* tr = &sT[c * DIM];
        const float* wr = &Wk[(h24 + d) * DIM];
        float acc = 0.f;
        for (int j = 0; j < DIM; ++j) acc += tr[j] * wr[j];
        sS[c * 25 + d] = acc;
    }
    __syncthreads();
    // ||q_c||^2 via Cc (reuse sT)
    for (int idx = tid; idx < DH * DIM; idx += 256) {
        int c = idx / DIM, j = idx % DIM;
        const float* wr = &Wq[(h24 + c) * DIM];
        float acc = 0.f;
        for (int ci = 0; ci < DIM; ++ci) acc += wr[ci] * Ccb[ci * DIM + j];
        sT[idx] = acc;
    }
    __syncthreads();
    if (tid < DH) {
        const float* wr = &Wq[(h24 + tid) * DIM];
        const float* tr = &sT[tid * DIM];
        float acc = 0.f;
        for (int j = 0; j < DIM; ++j) acc += tr[j] * wr[j];
        sNq[tid] = 1.0f / fmaxf(sqrtf(fmaxf(acc, 0.f)), EPS);
    }
    __syncthreads();
    // ||k_d||^2 via Ci (reuse sT)
    for (int idx = tid; idx < DH * DIM; idx += 256) {
        int c = idx / DIM, j = idx % DIM;
        const float* wr = &Wk[(h24 + c) * DIM];
        float acc = 0.f;
        for (int ci = 0; ci < DIM; ++ci) acc += wr[ci] * Cib[ci * DIM + j];
        sT[idx] = acc;
    }
    __syncthreads();
    if (tid < DH) {
        const float* wr = &Wk[(h24 + tid) * DIM];
        const float* tr = &sT[tid * DIM];
        float acc = 0.f;
        for (int j = 0; j < DIM; ++j) acc += tr[j] * wr[j];
        sNk[tid] = 1.0f / fmaxf(sqrtf(fmaxf(acc, 0.f)), EPS);
    }
    __syncthreads();
    // softmax per row
    if (tid < DH) {
        const int c = tid;
        const float iq = sNq[c];
        float row[DH];
        float mx = -INFINITY;
        #pragma unroll
        for (int d = 0; d < DH; ++d) {
            float s = sS[c * 25 + d] * iq * sNk[d];
            row[d] = s;
            mx = fmaxf(mx, s);
        }
        float sum = 0.f;
        #pragma unroll
        for (int d = 0; d < DH; ++d) {
            float e = __expf(row[d] - mx);
            row[d] = e;
            sum += e;
        }
        const float inv = 1.0f / sum;
        #pragma unroll
        for (int d = 0; d < DH; ++d) sP[c * 25 + d] = row[d] * inv;
    }
    __syncthreads();
    // Weff[b][o][h24+d] = sum_c Wo[o][h24+c] * P[c][d]
    float* Wb = Weff + (size_t)b * DIM * DIM;
    for (int idx = tid; idx < DIM * DH; idx += 256) {
        int o = idx / DH, d = idx % DH;
        const float* wr = &Wo[o * DIM + h24];
        float acc = 0.f;
        #pragma unroll
        for (int c = 0; c < DH; ++c) acc += wr[c] * sP[c * 25 + d];
        Wb[o * DIM + h24 + d] = acc;
    }
}

// ============================================================
// Kernel 3: W2[b] = Weff[b] * Wv  (192x192x192, tiny)
// ============================================================
__global__ __launch_bounds__(256)
void combine_kernel(const float* __restrict__ Weff,
                    const float* __restrict__ Wv,
                    float* __restrict__ W2)
{
    const int b = blockIdx.x;
    const float* Wb  = Weff + (size_t)b * DIM * DIM;
    float*       W2b = W2   + (size_t)b * DIM * DIM;
    for (int idx = threadIdx.x; idx < DIM * DIM; idx += 256) {
        int o = idx / DIM, ci = idx % DIM;
        float acc = 0.f;
        for (int d = 0; d < DIM; ++d) acc += Wb[o * DIM + d] * Wv[d * DIM + ci];
        W2b[idx] = acc;
    }
}

// ============================================================
// Kernel 4: out[b] = W2[b] * input[b]   (192 x 16384, K=192), WMMA.
// grid (256, 8), block 256 = 8 waves: wave = (ntile 0..3) x (m-half 0..1),
// each wave accumulates 6 m-tiles for one 16-wide n strip.
// B fragments are fully coalesced (n = contiguous hw axis).
// ============================================================
__global__ __launch_bounds__(256)
void final_gemm(const float* __restrict__ W2,
                const float* __restrict__ X,
                float* __restrict__ Out)
{
    const int b    = blockIdx.y;
    const int tid  = threadIdx.x;
    const int lane = tid & 31;
    const int wave = tid >> 5;
    const int nt   = wave >> 1;          // 0..3
    const int mh   = wave & 1;           // 0..1  (m halves of 96)
    const int n0   = blockIdx.x * 64 + nt * 16;

    const float* Wb = W2 + (size_t)b * DIM * DIM;
    const float* Xb = X  + (size_t)b * DIM * HW;
    float*       Ob = Out + (size_t)b * DIM * HW;

    const int ln = lane & 15;
    const int kh = lane >> 4;

    v8f acc[6] = {};

    for (int k = 0; k < DIM; k += 4) {
        v2f bfrag;
        bfrag.x = Xb[(size_t)(k + kh) * HW + n0 + ln];
        bfrag.y = Xb[(size_t)(k + 2 + kh) * HW + n0 + ln];
        #pragma unroll
        for (int mt = 0; mt < 6; ++mt) {
            const int m = mh * 96 + mt * 16 + ln;
            v2f afrag = *reinterpret_cast<const v2f*>(&Wb[m * DIM + k + kh * 2]);
            acc[mt] = __builtin_amdgcn_wmma_f32_16x16x4_f32(
                false, afrag, false, bfrag, (short)0, acc[mt], false, false);
        }
    }
    #pragma unroll
    for (int mt = 0; mt < 6; ++mt) {
        const int rbase = mh * 96 + mt * 16 + (kh << 3);
        #pragma unroll
        for (int r = 0; r < 8; ++r)
            Ob[(size_t)(rbase + r) * HW + n0 + ln] = acc[mt][r];
    }
}

// ============================================================
// Launch
// ============================================================
extern "C" void kernel_launch(void* const* d_in, const int* in_sizes, int n_in,
                              void* d_out, int out_size, void* d_ws, size_t ws_size,
                              hipStream_t stream) {
    const float* input = (const float*)d_in[0];
    const float* color = (const float*)d_in[1];
    const float* Wq    = (const float*)d_in[2];
    const float* Wk    = (const float*)d_in[3];
    const float* Wv    = (const float*)d_in[4];
    const float* Wo    = (const float*)d_in[5];
    float*       out   = (float*)d_out;

    // workspace: 5 per-batch 192x192 matrix sets (~5.9 MB total)
    float* ws   = (float*)d_ws;
    const size_t MAT = (size_t)DIM * DIM;  // 36864
    float* Cc   = ws;                 // color * color^T
    float* Ci   = Cc + NB * MAT;      // input * input^T
    float* M    = Ci + NB * MAT;      // color * input^T
    float* Weff = M  + NB * MAT;      // Wo folded with softmax probs
    float* W2   = Weff + NB * MAT;    // Weff * Wv

    cov_kernel<<<dim3(18, NB), 256, 0, stream>>>(color, color, Cc);
    cov_kernel<<<dim3(18, NB), 256, 0, stream>>>(input, input, Ci);
    cov_kernel<<<dim3(18, NB), 256, 0, stream>>>(color, input, M);
    head_kernel<<<dim3(NHEAD, NB), 256, 0, stream>>>(M, Cc, Ci, Wq, Wk, Wo, Weff);
    combine_kernel<<<NB, 256, 0, stream>>>(Weff, Wv, W2);
    final_gemm<<<dim3(HW / 64, NB), 256, 0, stream>>>(W2, input, out);
}